// USCLloss_17471926960647
// MI455X (gfx1250) — compile-verified
//
#include <hip/hip_runtime.h>
#include <math.h>

// ---------------------------------------------------------------------------
// USCL loss (SimCLR NT-Xent + supervised CE) for MI455X / gfx1250, wave32.
// B=4096, D=2048, P=128, C=10, TEMP=0.5, LAMBDA=0.2, EPS=1e-8
// Matrix work: V_WMMA_F32_16X16X4_F32 (fp32 WMMA, 16x16 tiles, K=4).
// All LDS staging uses b128 (float4) batched copies.
// ---------------------------------------------------------------------------

typedef float v2f_t __attribute__((ext_vector_type(2)));
typedef float v8f_t __attribute__((ext_vector_type(8)));

#define NB    4096
#define TWOB  8192
#define DD    2048
#define PP    128
#define CC    10
#define INV_TEMP 2.0f
#define LAMBDA   0.2f
#define EPS_F    1e-8f

static __device__ __forceinline__ v8f_t wmma_f32(v2f_t a, v2f_t b, v8f_t c) {
  // 8 args: (neg_a, A, neg_b, B, c_mod, C, reuse_a, reuse_b)
  return __builtin_amdgcn_wmma_f32_16x16x4_f32(false, a, false, b, (short)0, c,
                                               false, false);
}

// ---------------------------------------------------------------------------
// Kernel 1: one pass over xi/xj. Waves 0..7 -> proj tile (16 rows x 128 cols),
// wave 8 -> digits tile (16 rows x 16 cols, cols 10..15 zero-padded B).
// Epilogue: row L2 norms in LDS, write normalized rows (nhat).
// grid = 512 (one block per 16-row strip of the concatenated [xi;xj]).
// ---------------------------------------------------------------------------
__global__ __launch_bounds__(288) void k_proj_digits(
    const float* __restrict__ xi, const float* __restrict__ xj,
    const float* __restrict__ Wp, const float* __restrict__ Wc,
    float* __restrict__ nhat, float* __restrict__ digits)
{
  __shared__ __align__(16) float As[16 * 132];  // staged A chunk / proj tile
  __shared__ float rnorm[16];

  const int strip   = blockIdx.x;          // 0..511
  const int rowbase = strip * 16;          // global row in [0, 8192)
  const float* X    = (rowbase < NB) ? xi : xj;
  const int xrow    = (rowbase < NB) ? rowbase : (rowbase - NB);

  const int tid  = threadIdx.x;
  const int wave = tid >> 5;               // 0..8
  const int lane = tid & 31;
  const int m16  = lane & 15;
  const int koff = (lane < 16) ? 0 : 2;
  const int col  = wave * 16 + m16;        // proj column for waves 0..7
  const bool isdig = (wave == 8);

  const float4* Xv = reinterpret_cast<const float4*>(X);

  // precompute staging indices (tile = 512 float4: r = idx>>5, c4 = idx&31)
  const int s_r0 = tid >> 5,          s_c0 = tid & 31;           // idx = tid
  const int s_r1 = (tid + 256) >> 5,  s_c1 = (tid + 256) & 31;   // idx = tid+256

  v8f_t acc = {};

  for (int K0 = 0; K0 < DD; K0 += 128) {
    __syncthreads();
    if (tid < 256) {
      // batched b128 copy: 2 float4 per thread, loads clustered before stores
      float4 v0 = Xv[(size_t)(xrow + s_r0) * (DD / 4) + (K0 >> 2) + s_c0];
      float4 v1 = Xv[(size_t)(xrow + s_r1) * (DD / 4) + (K0 >> 2) + s_c1];
      *reinterpret_cast<float4*>(&As[s_r0 * 132 + s_c0 * 4]) = v0;
      *reinterpret_cast<float4*>(&As[s_r1 * 132 + s_c1 * 4]) = v1;
    }
    __syncthreads();

    if (!isdig) {
      #pragma unroll
      for (int kk = 0; kk < 128; kk += 4) {
        v2f_t a, b;
        a.x = As[m16 * 132 + kk + koff];
        a.y = As[m16 * 132 + kk + koff + 1];
        b.x = Wp[(size_t)(K0 + kk + koff) * PP + col];
        b.y = Wp[(size_t)(K0 + kk + koff + 1) * PP + col];
        acc = wmma_f32(a, b, acc);
      }
    } else {
      #pragma unroll
      for (int kk = 0; kk < 128; kk += 4) {
        v2f_t a, b;
        a.x = As[m16 * 132 + kk + koff];
        a.y = As[m16 * 132 + kk + koff + 1];
        b.x = (m16 < CC) ? Wc[(size_t)(K0 + kk + koff) * CC + m16] : 0.0f;
        b.y = (m16 < CC) ? Wc[(size_t)(K0 + kk + koff + 1) * CC + m16] : 0.0f;
        acc = wmma_f32(a, b, acc);
      }
    }
  }

  __syncthreads();
  if (!isdig) {
    // scatter proj tile into LDS (reuse As as P[16][132])
    #pragma unroll
    for (int vr = 0; vr < 8; ++vr) {
      int m = vr + ((lane < 16) ? 0 : 8);
      As[m * 132 + col] = acc[vr];
    }
  } else {
    // write digits tile (16 cols, padded; cols >= 10 are exact zeros)
    #pragma unroll
    for (int vr = 0; vr < 8; ++vr) {
      int m = vr + ((lane < 16) ? 0 : 8);
      digits[(size_t)(rowbase + m) * 16 + m16] = acc[vr];
    }
  }
  __syncthreads();

  if (tid < 16) {
    float s = 0.0f;
    for (int c = 0; c < PP; ++c) { float v = As[tid * 132 + c]; s += v * v; }
    rnorm[tid] = 1.0f / sqrtf((s > 0.0f) ? s : 1e-30f);
  }
  __syncthreads();

  if (tid < 256) {
    float4* nv = reinterpret_cast<float4*>(nhat);
    float4 v0 = *reinterpret_cast<const float4*>(&As[s_r0 * 132 + s_c0 * 4]);
    float4 v1 = *reinterpret_cast<const float4*>(&As[s_r1 * 132 + s_c1 * 4]);
    float r0 = rnorm[s_r0], r1 = rnorm[s_r1];
    v0.x *= r0; v0.y *= r0; v0.z *= r0; v0.w *= r0;
    v1.x *= r1; v1.y *= r1; v1.z *= r1; v1.w *= r1;
    nv[(size_t)(rowbase + s_r0) * (PP / 4) + s_c0] = v0;
    nv[(size_t)(rowbase + s_r1) * (PP / 4) + s_c1] = v1;
  }
}

// ---------------------------------------------------------------------------
// Kernel 2: fused S = exp(2 * nhat @ nhat^T), zero diagonal, row-sum only.
// grid = 512 row-strips, 4 waves/block; wave w handles col tiles w, w+4, ...
// ---------------------------------------------------------------------------
__global__ __launch_bounds__(128) void k_rowsum(
    const float* __restrict__ nhat, float* __restrict__ row_sum)
{
  __shared__ __align__(16) float Anh[16 * 132];
  __shared__ __align__(16) float Bs[4][16 * 132];
  __shared__ float rowpart[4][16];

  const int strip   = blockIdx.x;
  const int rowbase = strip * 16;
  const int tid  = threadIdx.x;
  const int wave = tid >> 5;               // 0..3
  const int lane = tid & 31;
  const int m16  = lane & 15;
  const int koff = (lane < 16) ? 0 : 2;
  const int mrow = ((lane < 16) ? 0 : 8);  // row offset of this lane's group

  const float4* nv = reinterpret_cast<const float4*>(nhat);

  // stage A strip: 512 float4, 4 per thread, batched
  {
    float4 t[4];
    #pragma unroll
    for (int q = 0; q < 4; ++q) {
      int idx = tid + q * 128;
      t[q] = nv[(size_t)(rowbase + (idx >> 5)) * 32 + (idx & 31)];
    }
    #pragma unroll
    for (int q = 0; q < 4; ++q) {
      int idx = tid + q * 128;
      *reinterpret_cast<float4*>(&Anh[(idx >> 5) * 132 + (idx & 31) * 4]) = t[q];
    }
  }
  __syncthreads();

  float racc[8];
  #pragma unroll
  for (int vr = 0; vr < 8; ++vr) racc[vr] = 0.0f;

  float* bs = &Bs[wave][0];
  for (int ct = wave; ct < TWOB / 16; ct += 4) {   // 128 iters per wave
    // stage B col-tile (16 rows of nhat): 512 float4, 16/lane, batched x4
    const float4* src = nv + (size_t)ct * 16 * 32;
    #pragma unroll
    for (int qq = 0; qq < 16; qq += 4) {
      float4 t0 = src[(qq + 0) * 32 + lane];
      float4 t1 = src[(qq + 1) * 32 + lane];
      float4 t2 = src[(qq + 2) * 32 + lane];
      float4 t3 = src[(qq + 3) * 32 + lane];
      *reinterpret_cast<float4*>(&bs[(qq + 0) * 132 + lane * 4]) = t0;
      *reinterpret_cast<float4*>(&bs[(qq + 1) * 132 + lane * 4]) = t1;
      *reinterpret_cast<float4*>(&bs[(qq + 2) * 132 + lane * 4]) = t2;
      *reinterpret_cast<float4*>(&bs[(qq + 3) * 132 + lane * 4]) = t3;
    }
    __syncthreads();

    v8f_t acc = {};
    #pragma unroll
    for (int kk = 0; kk < PP; kk += 4) {
      v2f_t a, b;
      a.x = Anh[m16 * 132 + kk + koff];
      a.y = Anh[m16 * 132 + kk + koff + 1];
      b.x = bs[m16 * 132 + kk + koff];
      b.y = bs[m16 * 132 + kk + koff + 1];
      acc = wmma_f32(a, b, acc);
    }

    const int gc = ct * 16 + m16;
    #pragma unroll
    for (int vr = 0; vr < 8; ++vr) {
      int gr = rowbase + vr + mrow;
      float e = __expf(acc[vr] * INV_TEMP);
      racc[vr] += (gr == gc) ? 0.0f : e;     // (1 - eye) mask
    }
    __syncthreads();
  }

  // reduce across the 16 lanes of each half-wave (N dimension)
  #pragma unroll
  for (int vr = 0; vr < 8; ++vr) {
    float v = racc[vr];
    v += __shfl_xor(v, 1, 32);
    v += __shfl_xor(v, 2, 32);
    v += __shfl_xor(v, 4, 32);
    v += __shfl_xor(v, 8, 32);
    racc[vr] = v;
  }
  if (lane == 0 || lane == 16) {
    #pragma unroll
    for (int vr = 0; vr < 8; ++vr) rowpart[wave][mrow + vr] = racc[vr];
  }
  __syncthreads();
  if (tid < 16) {
    float s = 0.0f;
    for (int w = 0; w < 4; ++w) s += rowpart[w][tid];   // fixed order
    row_sum[rowbase + tid] = s;
  }
}

// ---------------------------------------------------------------------------
// Kernel 3: positive pairs pos[i] = exp(2 * dot(nhat_i, nhat_{i+B})).
// One wave per i; one float4 per lane covers the 128-dim row.
// ---------------------------------------------------------------------------
__global__ __launch_bounds__(256) void k_pos(
    const float* __restrict__ nhat, float* __restrict__ pos)
{
  const int wave = threadIdx.x >> 5;
  const int lane = threadIdx.x & 31;
  const int i = blockIdx.x * 8 + wave;     // 512 blocks * 8 waves = 4096
  const float4* nv = reinterpret_cast<const float4*>(nhat);
  float4 a = nv[(size_t)i * 32 + lane];
  float4 b = nv[(size_t)(i + NB) * 32 + lane];
  float s = a.x * b.x + a.y * b.y + a.z * b.z + a.w * b.w;
  s += __shfl_xor(s, 16, 32);
  s += __shfl_xor(s, 8, 32);
  s += __shfl_xor(s, 4, 32);
  s += __shfl_xor(s, 2, 32);
  s += __shfl_xor(s, 1, 32);
  if (lane == 0) pos[i] = __expf(s * INV_TEMP);
}

// ---------------------------------------------------------------------------
// Kernel 4: final scalar. NT-Xent terms + CE over digits; deterministic
// tree reduction in one block.
// ---------------------------------------------------------------------------
__global__ __launch_bounds__(256) void k_final(
    const float* __restrict__ row_sum, const float* __restrict__ pos,
    const float* __restrict__ digits, const int* __restrict__ y,
    float* __restrict__ out)
{
  __shared__ float red[256];
  const int tid = threadIdx.x;

  float u = 0.0f;
  for (int i = tid; i < NB; i += 256) {
    float p = pos[i];
    u -= logf(p / row_sum[i] + EPS_F);
    u -= logf(p / row_sum[i + NB] + EPS_F);
  }

  float sp = 0.0f;
  for (int r = tid; r < TWOB; r += 256) {
    const float* l = &digits[(size_t)r * 16];
    float mx = l[0];
    #pragma unroll
    for (int c = 1; c < CC; ++c) mx = fmaxf(mx, l[c]);
    float se = 0.0f;
    #pragma unroll
    for (int c = 0; c < CC; ++c) se += __expf(l[c] - mx);
    float lse = logf(se) + mx;
    int lab = y[r & (NB - 1)];
    sp += lse - l[lab];
  }

  red[tid] = u * (1.0f / (float)TWOB) + LAMBDA * sp * (1.0f / (float)NB);
  __syncthreads();
  for (int s2 = 128; s2 > 0; s2 >>= 1) {
    if (tid < s2) red[tid] += red[tid + s2];
    __syncthreads();
  }
  if (tid == 0) out[0] = red[0];
}

// ---------------------------------------------------------------------------
extern "C" void kernel_launch(void* const* d_in, const int* in_sizes, int n_in,
                              void* d_out, int out_size, void* d_ws, size_t ws_size,
                              hipStream_t stream)
{
  (void)in_sizes; (void)n_in; (void)out_size; (void)ws_size;
  // inputs: x (unused), xi, xj, y, W_proj, W_cls
  const float* xi = (const float*)d_in[1];
  const float* xj = (const float*)d_in[2];
  const int*   y  = (const int*)d_in[3];
  const float* Wp = (const float*)d_in[4];
  const float* Wc = (const float*)d_in[5];

  float* ws      = (float*)d_ws;
  float* nhat    = ws;                               // 8192*128 = 1,048,576 f
  float* digits  = nhat + (size_t)TWOB * PP;         // 8192*16  =   131,072 f
  float* rs      = digits + (size_t)TWOB * 16;       // 8192 f
  float* pos     = rs + TWOB;                        // 4096 f

  k_proj_digits<<<TWOB / 16, 288, 0, stream>>>(xi, xj, Wp, Wc, nhat, digits);
  k_rowsum     <<<TWOB / 16, 128, 0, stream>>>(nhat, rs);
  k_pos        <<<NB / 8,    256, 0, stream>>>(nhat, pos);
  k_final      <<<1,         256, 0, stream>>>(rs, pos, digits, y, (float*)d_out);
}